// HeatmapHead_33036888441041
// MI455X (gfx1250) — compile-verified
//
#include <hip/hip_runtime.h>
#include <stdint.h>

// ---------------------------------------------------------------------------
// HeatmapHead loss + heatmap for MI455X (gfx1250).
//
// Roofline: 16 MB predict read + 8 MB heatmap write = 24 MB -> ~1.0 us at
// 23.3 TB/s HBM. Per-row scalar culling of the 160 rotated boxes keeps VALU
// work far below that floor, so the kernel is HBM-bound. No GEMM -> no WMMA;
// CDNA5 paths used instead: async global->LDS copy (+ s_wait_asynccnt),
// wave32 shuffles, nontemporal streaming stores, native f64/u32 atomics.
// ---------------------------------------------------------------------------

#define BB 8
#define CC 2
#define HH 512
#define WW 512
#define MAXN 256
#define EPSV 1e-7f

typedef int v4i __attribute__((vector_size(16)));
typedef __attribute__((address_space(1))) v4i gv4i;   // global (__device__)
typedef __attribute__((address_space(3))) v4i lv4i;   // LDS (__shared__)
typedef __attribute__((address_space(3))) char lchar;

struct __align__(16) TgtD {
  int   bid, cid;
  float c, s;                  // cos/sin of angle
  float x3, x4, y3, y4;        // inner box in rotated coords
  float x1, x2, y1, y2;        // outer box in rotated coords
  float gylo, gyhi;            // pixel-space row AABB of outer box
  float pad0, pad1;
};                              // 64 bytes

// workspace layout (bytes):
//   [0..31]   4 x double  : sum(log conf | obj) c0,c1 ; sum(log(1-conf) | noobj) c0,c1
//   [32..47]  4 x uint    : n_obj c0,c1 ; n_noobj c0,c1
//   [64.. ]   TgtD[N] table
#define WS_TAB_OFF 64

__global__ void hh_derive(const float* __restrict__ targets, int N, void* ws) {
  const int t = threadIdx.x;
  double*   dacc = (double*)ws;
  unsigned* uacc = (unsigned*)((char*)ws + 32);
  if (t < 4) { dacc[t] = 0.0; uacc[t] = 0u; }
  if (t >= N) return;

  const float* tr = targets + (size_t)t * 7;
  const float cx = tr[2] * 0.125f;
  const float cy = tr[3] * 0.125f;
  const float w  = tr[4] * 0.125f;
  const float h  = tr[5] * 0.125f;
  float s, c;
  __sincosf(tr[6], &s, &c);

  const float x  =  cx * c + cy * s;
  const float y  = -cx * s + cy * c;
  TgtD d;
  d.bid = (int)tr[0];
  d.cid = (int)tr[1];
  d.c = c; d.s = s;
  d.x3 = x - 0.5f * h;  d.x4 = x + 0.5f * h;
  d.y3 = y - 0.5f * w;  d.y4 = y + 0.5f * w;
  d.x1 = d.x3 - 0.5f;   d.x2 = d.x4 + 0.5f;
  d.y1 = d.y3 - 0.5f;   d.y2 = d.y4 + 0.5f;
  // pixel-space AABB of the outer box: center back-rotates to (cx, cy);
  // row half-extent = |s|*dx + |c|*dy with dx = h/2+0.5, dy = w/2+0.5.
  const float dx = 0.5f * h + 0.5f, dy = 0.5f * w + 0.5f;
  const float egy = fabsf(s) * dx + fabsf(c) * dy;
  d.gylo = cy - egy;
  d.gyhi = cy + egy;
  d.pad0 = 0.0f; d.pad1 = 0.0f;
  ((TgtD*)((char*)ws + WS_TAB_OFF))[t] = d;
}

__device__ __forceinline__ float sreadf(float v) {
  return __uint_as_float(__builtin_amdgcn_readfirstlane(__float_as_uint(v)));
}

__global__ void __launch_bounds__(256) hh_main(const float* __restrict__ predict,
                                               void* ws, float* out, int N) {
  __shared__ TgtD     tab[MAXN];
  __shared__ double   fa[4];
  __shared__ unsigned ua[4];

  const int tid = threadIdx.x;
  const int b   = blockIdx.x / HH;
  const int h   = blockIdx.x - b * HH;

  // ---- cooperative copy of the target table into LDS (async on CDNA5) ----
  const int nv = N * 4;                         // number of 16B chunks
#if __has_builtin(__builtin_amdgcn_global_load_async_to_lds_b128)
  {
    v4i*   srcv  = (v4i*)((char*)ws + WS_TAB_OFF);
    lchar* lbase = (lchar*)(char*)tab;
    for (int i = tid; i < nv; i += 256) {
      __builtin_amdgcn_global_load_async_to_lds_b128(
          (gv4i*)(srcv + i), (lv4i*)(lbase + (size_t)i * 16), 0, 0);
    }
    asm volatile("s_wait_asynccnt 0" ::: "memory");
  }
#else
  {
    const float4* src = (const float4*)((const char*)ws + WS_TAB_OFF);
    for (int i = tid; i < nv; i += 256) ((float4*)tab)[i] = src[i];
  }
#endif
  if (tid < 4) { fa[tid] = 0.0; ua[tid] = 0u; }
  __syncthreads();

  // ---- per-pixel box membership, scalar-culled target loop ----
  // Each thread owns two ADJACENT pixels: w0 = 2*tid, w1 = 2*tid+1
  // (64-bit coalesced predict loads; vx/vy for pixel1 = pixel0 + (c, -s)).
  const float gy  = sreadf((float)h + 0.5f);          // uniform -> SGPR
  const float gx0 = 2.0f * (float)tid + 0.5f;

  int ob00 = 0, ob01 = 0, ob10 = 0, ob11 = 0;   // obj  [pixel][class]
  int ou00 = 0, ou01 = 0, ou10 = 0, ou11 = 0;   // outer[pixel][class]

  for (int t = 0; t < N; ++t) {
    const int tb = __builtin_amdgcn_readfirstlane(tab[t].bid);
    if (tb != b) continue;                       // scalar skip
    const float lo = sreadf(tab[t].gylo);
    const float hi = sreadf(tab[t].gyhi);
    if (gy < lo || gy > hi) continue;            // scalar row cull

    const int   cid = __builtin_amdgcn_readfirstlane(tab[t].cid);
    const float c = tab[t].c,  s = tab[t].s;
    const float x3 = tab[t].x3, x4 = tab[t].x4, y3 = tab[t].y3, y4 = tab[t].y4;
    const float x1 = tab[t].x1, x2 = tab[t].x2, y1 = tab[t].y1, y2 = tab[t].y2;

    const float vx0 = fmaf(c, gx0,  s * gy);
    const float vy0 = fmaf(c, gy,  -s * gx0);
    const float vx1 = vx0 + c;                   // gx step of +1
    const float vy1 = vy0 - s;

    const int in0 = (vx0 >= x3) & (vx0 <= x4) & (vy0 >= y3) & (vy0 <= y4);
    const int o0  = (vx0 >= x1) & (vx0 <= x2) & (vy0 >= y1) & (vy0 <= y2);
    const int in1 = (vx1 >= x3) & (vx1 <= x4) & (vy1 >= y3) & (vy1 <= y4);
    const int o1  = (vx1 >= x1) & (vx1 <= x2) & (vy1 >= y1) & (vy1 <= y2);

    if (cid == 0) { ob00 |= in0; ou00 |= o0; ob10 |= in1; ou10 |= o1; }
    else          { ob01 |= in0; ou01 |= o0; ob11 |= in1; ou11 |= o1; }
  }

  // ---- sigmoid / heatmap / masked log accumulation ----
  const size_t hw = (size_t)HH * WW;
  const float* p0 = predict + (size_t)(b * CC) * hw + (size_t)h * WW;
  const float* p1 = p0 + hw;
  const float2 z0 = *(const float2*)(p0 + 2 * tid);   // class 0, pixels 0/1
  const float2 z1 = *(const float2*)(p1 + 2 * tid);   // class 1, pixels 0/1
  const float c00 = 1.0f / (1.0f + __expf(-z0.x));
  const float c01 = 1.0f / (1.0f + __expf(-z1.x));
  const float c10 = 1.0f / (1.0f + __expf(-z0.y));
  const float c11 = 1.0f / (1.0f + __expf(-z1.y));

  float* ho = out + 1 + (size_t)(b * HH + h) * WW + 2 * tid;
  __builtin_nontemporal_store(fmaxf(c00, c01), ho);
  __builtin_nontemporal_store(fmaxf(c10, c11), ho + 1);

  float sO0 = 0.f, sO1 = 0.f, sN0 = 0.f, sN1 = 0.f;
  int   nO0 = 0,   nO1 = 0,   nN0 = 0,   nN1 = 0;
  auto proc = [&](float conf, int isobj, int isout, float& so, int& co,
                  float& sn, int& cn) {
    const float cc = fminf(fmaxf(conf, EPSV), 1.0f - EPSV);
    if (isobj)  { so += __logf(cc);        co++; }
    if (!isout) { sn += __logf(1.0f - cc); cn++; }
  };
  proc(c00, ob00, ou00, sO0, nO0, sN0, nN0);
  proc(c01, ob01, ou01, sO1, nO1, sN1, nN1);
  proc(c10, ob10, ou10, sO0, nO0, sN0, nN0);
  proc(c11, ob11, ou11, sO1, nO1, sN1, nN1);

  // ---- wave32 reduction, then LDS + global atomics ----
  auto wredf = [](float v) { for (int m = 16; m; m >>= 1) v += __shfl_xor(v, m, 32); return v; };
  auto wredi = [](int v)   { for (int m = 16; m; m >>= 1) v += __shfl_xor(v, m, 32); return v; };
  sO0 = wredf(sO0); sO1 = wredf(sO1); sN0 = wredf(sN0); sN1 = wredf(sN1);
  nO0 = wredi(nO0); nO1 = wredi(nO1); nN0 = wredi(nN0); nN1 = wredi(nN1);

  if ((tid & 31) == 0) {
    atomicAdd(&fa[0], (double)sO0); atomicAdd(&fa[1], (double)sO1);
    atomicAdd(&fa[2], (double)sN0); atomicAdd(&fa[3], (double)sN1);
    atomicAdd(&ua[0], (unsigned)nO0); atomicAdd(&ua[1], (unsigned)nO1);
    atomicAdd(&ua[2], (unsigned)nN0); atomicAdd(&ua[3], (unsigned)nN1);
  }
  __syncthreads();
  if (tid == 0) {
    double*   gd = (double*)ws;
    unsigned* gu = (unsigned*)((char*)ws + 32);
    atomicAdd(&gd[0], fa[0]); atomicAdd(&gd[1], fa[1]);
    atomicAdd(&gd[2], fa[2]); atomicAdd(&gd[3], fa[3]);
    atomicAdd(&gu[0], ua[0]); atomicAdd(&gu[1], ua[1]);
    atomicAdd(&gu[2], ua[2]); atomicAdd(&gu[3], ua[3]);
  }
}

__global__ void hh_final(const void* ws, float* out) {
  const double*   d = (const double*)ws;
  const unsigned* u = (const unsigned*)((const char*)ws + 32);
  float loss = 0.0f;
  for (int c = 0; c < CC; ++c) {
    const unsigned no = u[c], nn = u[2 + c];
    const float lobj = (float)(-d[c]     / (double)(no > 0u ? no : 1u));
    const float lno  = (float)(-d[2 + c] / (double)(nn > 0u ? nn : 1u));
    if (no > 0u) loss += lobj + lno;   // NO_OBJ_SCALE == 1
  }
  out[0] = loss;
}

extern "C" void kernel_launch(void* const* d_in, const int* in_sizes, int n_in,
                              void* d_out, int out_size, void* d_ws, size_t ws_size,
                              hipStream_t stream) {
  const float* predict = (const float*)d_in[0];
  const float* targets = (const float*)d_in[1];
  float*       out     = (float*)d_out;
  int N = in_sizes[1] / 7;
  if (N > MAXN) N = MAXN;

  hh_derive<<<1, 256, 0, stream>>>(targets, N, d_ws);
  hh_main<<<BB * HH, 256, 0, stream>>>(predict, d_ws, out, N);
  hh_final<<<1, 1, 0, stream>>>(d_ws, out);
}